// LSTM_17721035063955
// MI455X (gfx1250) — compile-verified
//
#include <hip/hip_runtime.h>
#include <cmath>

// Problem constants (from the reference)
#define T_SEQ  512      // sequence length (xe.shape[0])
#define NBATCH 2048     // batch (only column NBATCH-1 reaches the output!)
#define E_DIM  50       // embedding dim
#define H_DIM  64       // hidden
#define G_DIM  256      // 4*H gates

typedef float v2f __attribute__((ext_vector_type(2)));
typedef float v8f __attribute__((ext_vector_type(8)));

#if defined(__HIP_DEVICE_COMPILE__)
#if !__has_builtin(__builtin_amdgcn_wmma_f32_16x16x4_f32)
#error "device pass: missing __builtin_amdgcn_wmma_f32_16x16x4_f32"
#endif
#endif

// ---------------------------------------------------------------------------
// K1: embedding gather (batch col 2047 only) + input projection GEMM.
//     P[t][g] = sum_e emb[x[t,2047]][e] * W_ih[g][e] + b_ih[g] + b_hh[g]
//     (512 x 52pad) @ (52pad x 256) via V_WMMA_F32_16X16X4_F32, one wave/tile.
// A 16x4 layout: lanes 0-15 hold K=0,1 ; lanes 16-31 hold K=2,3 (M = lane&15)
// B 4x16 layout (mirrored): VGPR j holds row K = j + 2*half, N = lane&15
// C/D: VGPR r -> M = r + 8*half, N = lane&15
//
// K handling: 12 unconditional K=4 steps cover e in [0,48) (max index 47).
// One peeled tail step loads e=48,49 unconditionally (always in-bounds) and
// masks half=1 lanes (which would be e=50,51) to zero via a multiply —
// no divergent branches, no exec churn around the loads.
// All A/B element addresses are even-float => 8B aligned => float2 loads.
// ---------------------------------------------------------------------------
__global__ void k1_xproj(const int* __restrict__ x,
                         const float* __restrict__ emb,
                         const float* __restrict__ W_ih,
                         const float* __restrict__ b_ih,
                         const float* __restrict__ b_hh,
                         float* __restrict__ P)
{
    const int lane = threadIdx.x;      // 0..31 (wave32, one wave per block)
    const int half = lane >> 4;        // 0 or 1
    const int l    = lane & 15;
    const int g0   = blockIdx.x * 16;  // gate tile
    const int t0   = blockIdx.y * 16;  // time tile

    const int tok = x[(t0 + l) * NBATCH + (NBATCH - 1)];
    const float* arow = emb  + (long)tok * E_DIM;        // A row (M = t0+l)
    const float* brow = W_ih + (long)(g0 + l) * E_DIM;   // B col source (N = g0+l)

    const float bias = b_ih[g0 + l] + b_hh[g0 + l];
    v8f c;
    #pragma unroll
    for (int r = 0; r < 8; ++r) c[r] = bias;

    #pragma unroll
    for (int e0 = 0; e0 < 48; e0 += 4) {   // 12 full K-steps, all in-bounds
        const int eA = e0 + 2 * half;
        const float2 av = *(const float2*)(arow + eA);
        const float2 bv = *(const float2*)(brow + eA);
        v2f a, b;
        a[0] = av.x; a[1] = av.y;
        b[0] = bv.x; b[1] = bv.y;
        c = __builtin_amdgcn_wmma_f32_16x16x4_f32(false, a, false, b,
                                                  (short)0, c, false, false);
    }
    {   // tail K-step: e = 48..51; columns 50,51 (half==1 lanes) are zero pad
        const float m = (half == 0) ? 1.0f : 0.0f;
        const float2 av = *(const float2*)(arow + 48);   // e=48,49: in-bounds
        const float2 bv = *(const float2*)(brow + 48);
        v2f a, b;
        a[0] = av.x * m; a[1] = av.y * m;
        b[0] = bv.x * m; b[1] = bv.y * m;
        c = __builtin_amdgcn_wmma_f32_16x16x4_f32(false, a, false, b,
                                                  (short)0, c, false, false);
    }

    #pragma unroll
    for (int r = 0; r < 8; ++r) {
        const int m = r + 8 * half;
        P[(t0 + m) * G_DIM + (g0 + l)] = c[r];
    }
}

// ---------------------------------------------------------------------------
// K2: sequential LSTM scan for the single surviving sequence.
//     One block, 256 threads (8 waves). Thread g owns gate row g of W_hh in
//     64 VGPRs; h[64] lives in LDS (broadcast reads); threads 0..63 own c.
// Gate order per jnp.split: i=[0,64) f=[64,128) g=[128,192) o=[192,256)
// ---------------------------------------------------------------------------
__global__ void k2_scan(const float* __restrict__ P,
                        const float* __restrict__ W_hh,
                        float* __restrict__ Hout)
{
    __shared__ __align__(16) float h[H_DIM];
    __shared__ float gact[G_DIM];
    const int g = threadIdx.x;   // 0..255

    float w[H_DIM];
    #pragma unroll
    for (int k = 0; k < H_DIM; ++k) w[k] = W_hh[g * H_DIM + k];

    float c = 0.0f;
    if (g < H_DIM) h[g] = 0.0f;
    __syncthreads();

    for (int t = 0; t < T_SEQ; ++t) {
        float acc = P[t * G_DIM + g];
        const float4* h4 = (const float4*)h;
        #pragma unroll
        for (int k4 = 0; k4 < H_DIM / 4; ++k4) {
            const float4 hv = h4[k4];
            acc = fmaf(w[4 * k4 + 0], hv.x, acc);
            acc = fmaf(w[4 * k4 + 1], hv.y, acc);
            acc = fmaf(w[4 * k4 + 2], hv.z, acc);
            acc = fmaf(w[4 * k4 + 3], hv.w, acc);
        }
        float act;
        if (g >= 2 * H_DIM && g < 3 * H_DIM) act = tanhf(acc);           // g-gate
        else                                 act = 1.0f / (1.0f + expf(-acc));
        gact[g] = act;
        __syncthreads();                      // gact ready; old-h reads done
        if (g < H_DIM) {
            c = gact[H_DIM + g] * c + gact[g] * gact[2 * H_DIM + g];
            const float hn = gact[3 * H_DIM + g] * tanhf(c);
            h[g] = hn;
            Hout[t * H_DIM + g] = hn;
        }
        __syncthreads();                      // h updated before next matvec
    }
}

// ---------------------------------------------------------------------------
// K3: fc1 GEMM + ReLU: h1 = relu(Hout(512x64) @ fc1_w^T(64x256) + fc1_b)
//     Same WMMA tiling as K1, K = 64 exactly (16 K-steps, no guards).
//     Rows are 256B apart and k is even => float2 (b64) loads.
// ---------------------------------------------------------------------------
__global__ void k3_fc1(const float* __restrict__ Hout,
                       const float* __restrict__ fc1_w,
                       const float* __restrict__ fc1_b,
                       float* __restrict__ h1)
{
    const int lane = threadIdx.x;
    const int half = lane >> 4;
    const int l    = lane & 15;
    const int g0   = blockIdx.x * 16;
    const int t0   = blockIdx.y * 16;

    const float* arow = Hout  + (long)(t0 + l) * H_DIM;
    const float* brow = fc1_w + (long)(g0 + l) * H_DIM;

    const float bias = fc1_b[g0 + l];
    v8f c;
    #pragma unroll
    for (int r = 0; r < 8; ++r) c[r] = bias;

    #pragma unroll
    for (int k0 = 0; k0 < H_DIM; k0 += 4) {
        const int k = k0 + 2 * half;
        const float2 av = *(const float2*)(arow + k);
        const float2 bv = *(const float2*)(brow + k);
        v2f a, b;
        a[0] = av.x; a[1] = av.y;
        b[0] = bv.x; b[1] = bv.y;
        c = __builtin_amdgcn_wmma_f32_16x16x4_f32(false, a, false, b,
                                                  (short)0, c, false, false);
    }

    #pragma unroll
    for (int r = 0; r < 8; ++r) {
        const int m = r + 8 * half;
        h1[(t0 + m) * G_DIM + (g0 + l)] = fmaxf(c[r], 0.0f);
    }
}

// ---------------------------------------------------------------------------
// K4: fc2 + sigmoid: out[t] = sigmoid(dot(h1[t], fc2_w) + fc2_b)
// ---------------------------------------------------------------------------
__global__ void k4_fc2(const float* __restrict__ h1,
                       const float* __restrict__ fc2_w,
                       const float* __restrict__ fc2_b,
                       float* __restrict__ out)
{
    __shared__ float red[G_DIM];
    const int t = blockIdx.x;
    const int j = threadIdx.x;
    red[j] = h1[t * G_DIM + j] * fc2_w[j];
    __syncthreads();
    #pragma unroll
    for (int s = G_DIM / 2; s > 0; s >>= 1) {
        if (j < s) red[j] += red[j + s];
        __syncthreads();
    }
    if (j == 0) out[t] = 1.0f / (1.0f + expf(-(red[0] + fc2_b[0])));
}

// ---------------------------------------------------------------------------
extern "C" void kernel_launch(void* const* d_in, const int* in_sizes, int n_in,
                              void* d_out, int out_size, void* d_ws, size_t ws_size,
                              hipStream_t stream)
{
    const int*   x     = (const int*)  d_in[0];
    const float* emb   = (const float*)d_in[1];
    const float* W_ih  = (const float*)d_in[2];
    const float* W_hh  = (const float*)d_in[3];
    const float* b_ih  = (const float*)d_in[4];
    const float* b_hh  = (const float*)d_in[5];
    const float* fc1_w = (const float*)d_in[6];
    const float* fc1_b = (const float*)d_in[7];
    const float* fc2_w = (const float*)d_in[8];
    const float* fc2_b = (const float*)d_in[9];
    float* out = (float*)d_out;

    float* P    = (float*)d_ws;                 // 512*256 f32 = 512 KB
    float* Hout = P + T_SEQ * G_DIM;            // 512*64  f32 = 128 KB
    float* h1   = P;                            // reuse P after the scan

    dim3 tiles(G_DIM / 16, T_SEQ / 16);         // 16 x 32 tiles, 1 wave each
    k1_xproj<<<tiles, 32, 0, stream>>>(x, emb, W_ih, b_ih, b_hh, P);
    k2_scan <<<1, G_DIM, 0, stream>>>(P, W_hh, Hout);
    k3_fc1  <<<tiles, 32, 0, stream>>>(Hout, fc1_w, fc1_b, h1);
    k4_fc2  <<<T_SEQ, G_DIM, 0, stream>>>(h1, fc2_w, fc2_b, out);
    (void)in_sizes; (void)n_in; (void)out_size; (void)ws_size;
}